// EnhancedGCN_79070347920043
// MI455X (gfx1250) — compile-verified
//
#include <hip/hip_runtime.h>
#include <hip/hip_bf16.h>

// ---------------------------------------------------------------------------
// EnhancedGCN on MI455X (gfx1250).
// Dense node-feature GEMMs use V_WMMA_F32_16X16X4_F32 (exact fp32 matrix op),
// guard-free: x is zero-padded K=5 -> K=8 so the WMMA k-loop has no bounds
// checks and EXEC stays all-ones (required by WMMA) with no exec-mask churn.
// Edge aggregation is L2-resident scatter-add via native global_atomic_add_f32
// (unsafeAtomicAdd -> no CAS loop), wave per edge, lane per channel.
// Workspace layout (floats):
//   dinv[N] | xp[N*8] | h[N*32] | h2[N*32] | agg[N*32] | stats[128] | pre[N]
//   total = 2*N + N*8 + 3*N*32 + 128 floats  (~42.4 MB for N=100000)
// ---------------------------------------------------------------------------

typedef __attribute__((ext_vector_type(2))) float v2f;
typedef __attribute__((ext_vector_type(8))) float v8f;

#define C32 32

// Native HW f32 atomic add (global_atomic_add_f32), not a CAS loop.
__device__ __forceinline__ void atomAddF32(float* p, float v) {
  unsafeAtomicAdd(p, v);
}

// ---------------- degree / dinv ----------------
__global__ void gcn_degree_kernel(const int* __restrict__ dst,
                                  int* __restrict__ degi, int E) {
  int stride = gridDim.x * blockDim.x;
  for (int e = blockIdx.x * blockDim.x + threadIdx.x; e < E; e += stride) {
    atomicAdd(&degi[dst[e]], 1);
  }
}

__global__ void gcn_dinv_kernel(float* __restrict__ dinv, int M) {
  int* degi = (int*)dinv;
  int stride = gridDim.x * blockDim.x;
  for (int i = blockIdx.x * blockDim.x + threadIdx.x; i < M; i += stride) {
    float d = (float)degi[i] + 1.0f;  // +1 self loop
    dinv[i] = rsqrtf(fmaxf(d, 1.0f));
  }
}

// ---------------- pad x[N,5] -> xp[N,8] (zero-filled) ----------------
__global__ void gcn_pad_x_kernel(const float* __restrict__ x,
                                 float* __restrict__ xp, int M) {
  int stride = gridDim.x * blockDim.x;
  int total = M * 8;
  for (int i = blockIdx.x * blockDim.x + threadIdx.x; i < total; i += stride) {
    int r = i >> 3;
    int c = i & 7;
    xp[i] = (c < 5) ? x[r * 5 + c] : 0.f;
  }
}

// ---------------- WMMA f32 GEMM: H[M,NC] = X[M,K] @ W[K,NC] ----------------
// Guard-free: covers full 16-row tiles only (tilesM = M/16); one wave per
// 16x16 output tile, K fully unrolled into a v_wmma_f32_16x16x4_f32 chain.
template <int K, int NC>
__global__ void gcn_gemm_wmma_kernel(const float* __restrict__ X,
                                     const float* __restrict__ W,
                                     float* __restrict__ H, int tilesM) {
  const int lane = threadIdx.x & 31;
  const int half = lane >> 4;     // 0: lanes 0-15, 1: lanes 16-31
  const int l15  = lane & 15;
  const int wavesPerBlock = blockDim.x >> 5;
  const int waveStride = gridDim.x * wavesPerBlock;
  const int tilesN = NC / 16;
  const int nTiles = tilesM * tilesN;

  for (int t = blockIdx.x * wavesPerBlock + (threadIdx.x >> 5); t < nTiles;
       t += waveStride) {
    const int tm = (t / tilesN) << 4;
    const int tn = (t % tilesN) << 4;
    const float* __restrict__ arow = X + (size_t)(tm + l15) * K;  // A row
    const float* __restrict__ bcol = W + tn + l15;                // B column

    v8f acc = {0.f, 0.f, 0.f, 0.f, 0.f, 0.f, 0.f, 0.f};
#pragma unroll
    for (int k = 0; k < K; k += 4) {
      const int kk = k + (half << 1);  // lanes 0-15: k+0/1, lanes 16-31: k+2/3
      v2f a = *(const v2f*)(arow + kk);          // aligned b64 load
      v2f b;
      b.x = bcol[kk * NC];
      b.y = bcol[(kk + 1) * NC];
      acc = __builtin_amdgcn_wmma_f32_16x16x4_f32(
          /*neg_a=*/false, a, /*neg_b=*/false, b,
          /*c_mod=*/(short)0, acc, /*reuse_a=*/false, /*reuse_b=*/false);
    }
    float* __restrict__ hout = H + (size_t)(tm + (half << 3)) * NC + tn + l15;
#pragma unroll
    for (int i = 0; i < 8; ++i) hout[i * NC] = acc[i];  // unconditional stores
  }
}

// scalar tail for rows [rowBeg, M): one thread per (row, col)
template <int K, int NC>
__global__ void gcn_gemm_tail_kernel(const float* __restrict__ X,
                                     const float* __restrict__ W,
                                     float* __restrict__ H, int rowBeg, int M) {
  int stride = gridDim.x * blockDim.x;
  int total = (M - rowBeg) * NC;
  for (int i = blockIdx.x * blockDim.x + threadIdx.x; i < total; i += stride) {
    int r = rowBeg + i / NC;
    int c = i % NC;
    float acc = 0.f;
    for (int k = 0; k < K; ++k) acc += X[(size_t)r * K + k] * W[k * NC + c];
    H[(size_t)r * NC + c] = acc;
  }
}

// ---------------- edge aggregation, 32 channels ----------------
// One wave per edge, lane = channel: agg[dst][c] += h[src][c]*dinv[src]*dinv[dst]
__global__ void gcn_aggregate32_kernel(const float* __restrict__ h,
                                       const float* __restrict__ dinv,
                                       const int* __restrict__ src,
                                       const int* __restrict__ dst,
                                       float* __restrict__ agg, int E) {
  const int lane = threadIdx.x & 31;
  const int wavesPerBlock = blockDim.x >> 5;
  const int stride = gridDim.x * wavesPerBlock;
  for (int e = blockIdx.x * wavesPerBlock + (threadIdx.x >> 5); e < E;
       e += stride) {
    __builtin_prefetch(&src[e + stride], 0, 1);  // global_prefetch_b8
    __builtin_prefetch(&dst[e + stride], 0, 1);
    const int s = src[e];
    const int d = dst[e];
    const float nrm = dinv[s] * dinv[d];
    atomAddF32(&agg[d * C32 + lane], h[s * C32 + lane] * nrm);
  }
}

// self-loop contribution: agg[i][c] += h[i][c]*dinv[i]^2 (non-atomic; stream-ordered)
__global__ void gcn_selfloop32_kernel(const float* __restrict__ h,
                                      const float* __restrict__ dinv,
                                      float* __restrict__ agg, int M) {
  const int lane = threadIdx.x & 31;
  const int wavesPerBlock = blockDim.x >> 5;
  const int stride = gridDim.x * wavesPerBlock;
  for (int i = blockIdx.x * wavesPerBlock + (threadIdx.x >> 5); i < M;
       i += stride) {
    const float di = dinv[i];
    agg[i * C32 + lane] += h[i * C32 + lane] * (di * di);
  }
}

// ---------------- batch-norm stats (sum, sumsq per channel) ----------------
// Requires blockDim.x == 256; stride is a multiple of 32 so channel == lane.
__global__ void gcn_bn_stats_kernel(const float* __restrict__ xv,
                                    float* __restrict__ stats, int total) {
  __shared__ float ssum[256];
  __shared__ float ssq[256];
  const int tid = threadIdx.x;
  const int stride = gridDim.x * blockDim.x;
  float a = 0.f, b = 0.f;
  for (int i = blockIdx.x * blockDim.x + tid; i < total; i += stride) {
    const float v = xv[i];
    a += v;
    b += v * v;
  }
  ssum[tid] = a;
  ssq[tid] = b;
  __syncthreads();
  if (tid < 32) {
    float sa = 0.f, sb = 0.f;
#pragma unroll
    for (int w = 0; w < 8; ++w) {
      sa += ssum[w * 32 + tid];
      sb += ssq[w * 32 + tid];
    }
    atomAddF32(&stats[tid], sa);        // per-channel sum
    atomAddF32(&stats[32 + tid], sb);   // per-channel sumsq
  }
}

// 1 block, 32 threads: stats[64+c]=scale, stats[96+c]=shift
__global__ void gcn_bn_finalize_kernel(float* __restrict__ stats,
                                       const float* __restrict__ gamma,
                                       const float* __restrict__ beta, int n) {
  const int c = threadIdx.x;
  const float invn = 1.0f / (float)n;
  const float mean = stats[c] * invn;
  const float var = fmaxf(stats[32 + c] * invn - mean * mean, 0.f);
  const float scl = gamma[c] * rsqrtf(var + 1e-5f);
  stats[64 + c] = scl;
  stats[96 + c] = beta[c] - mean * scl;
}

// out[i] = relu(in[i]*scale[c] + shift[c]),  c = i & 31
__global__ void gcn_bn_relu_kernel(const float* __restrict__ in,
                                   const float* __restrict__ stats,
                                   float* __restrict__ outv, int total) {
  const int stride = gridDim.x * blockDim.x;
  for (int i = blockIdx.x * blockDim.x + threadIdx.x; i < total; i += stride) {
    const int c = i & 31;
    outv[i] = fmaxf(fmaf(in[i], stats[64 + c], stats[96 + c]), 0.f);
  }
}

// ---------------- layer 3: pre[i] = h[i,:] . W3  (wave32 shuffle reduce) ----
__global__ void gcn_gemv32_kernel(const float* __restrict__ H,
                                  const float* __restrict__ W3,
                                  float* __restrict__ pre, int M) {
  const int lane = threadIdx.x & 31;
  const int wavesPerBlock = blockDim.x >> 5;
  const int stride = gridDim.x * wavesPerBlock;
  const float w = W3[lane];
  for (int i = blockIdx.x * wavesPerBlock + (threadIdx.x >> 5); i < M;
       i += stride) {
    float p = H[i * C32 + lane] * w;
#pragma unroll
    for (int off = 16; off > 0; off >>= 1) p += __shfl_xor(p, off, 32);
    if (lane == 0) pre[i] = p;
  }
}

// scalar edge aggregation (C=1)
__global__ void gcn_aggregate1_kernel(const float* __restrict__ pre,
                                      const float* __restrict__ dinv,
                                      const int* __restrict__ src,
                                      const int* __restrict__ dst,
                                      float* __restrict__ out, int E) {
  const int stride = gridDim.x * blockDim.x;
  for (int e = blockIdx.x * blockDim.x + threadIdx.x; e < E; e += stride) {
    __builtin_prefetch(&src[e + stride], 0, 1);
    __builtin_prefetch(&dst[e + stride], 0, 1);
    const int s = src[e];
    const int d = dst[e];
    atomAddF32(&out[d], pre[s] * dinv[s] * dinv[d]);
  }
}

__global__ void gcn_selfloop1_kernel(const float* __restrict__ pre,
                                     const float* __restrict__ dinv,
                                     const float* __restrict__ b3,
                                     float* __restrict__ out, int M) {
  const int stride = gridDim.x * blockDim.x;
  const float bias = b3[0];
  for (int i = blockIdx.x * blockDim.x + threadIdx.x; i < M; i += stride) {
    const float di = dinv[i];
    out[i] += pre[i] * (di * di) + bias;
  }
}

// ---------------------------------------------------------------------------
extern "C" void kernel_launch(void* const* d_in, const int* in_sizes, int n_in,
                              void* d_out, int out_size, void* d_ws,
                              size_t ws_size, hipStream_t stream) {
  const float* x      = (const float*)d_in[0];
  const int*   ei     = (const int*)d_in[1];   // jnp default: int32 (x64 disabled)
  const float* W1     = (const float*)d_in[2];
  const float* gamma1 = (const float*)d_in[4];
  const float* beta1  = (const float*)d_in[5];
  const float* W2     = (const float*)d_in[6];
  const float* gamma2 = (const float*)d_in[8];
  const float* beta2  = (const float*)d_in[9];
  const float* W3     = (const float*)d_in[10];
  const float* b3     = (const float*)d_in[11];
  float* out = (float*)d_out;

  const int N = in_sizes[0] / 5;   // 100000
  const int E = in_sizes[1] / 2;   // 6400000
  const int* src = ei;
  const int* dst = ei + E;

  // workspace carve-up
  float* dinv  = (float*)d_ws;
  float* xp    = dinv + N;              // padded x [N,8]
  float* h     = xp + (size_t)N * 8;    // activations / gemm input
  float* h2    = h + (size_t)N * C32;   // gemm output
  float* agg   = h2 + (size_t)N * C32;
  float* stats = agg + (size_t)N * C32; // [0,32)=sum [32,64)=sq [64,96)=scale [96,128)=shift
  float* pre   = stats + 128;

  const int blk = 256;
  const int edgeBlocks = 4096;   // grid-stride
  const int mapBlocks  = 2048;
  const int waveEdgeBlocks = 8192;              // 8 waves/block -> 64k edges/pass
  const int nodeWaveBlocks = (N + 7) / 8;       // wave per node
  const int tilesM = N >> 4;                    // full 16-row tiles
  const int remRows = N & 15;                   // tail rows (0 for N=100000)
  const int tiles32 = tilesM * (C32 / 16);
  const int gemmBlocks = (tiles32 + 7) / 8;     // 8 waves/block, tile per wave

  // ---- degrees ----
  hipMemsetAsync(dinv, 0, (size_t)N * sizeof(float), stream);
  gcn_degree_kernel<<<edgeBlocks, blk, 0, stream>>>(dst, (int*)dinv, E);
  gcn_dinv_kernel<<<mapBlocks, blk, 0, stream>>>(dinv, N);

  // ---- layer 1: conv(x, W1) -> BN -> ReLU  (bias cancels inside BN) ----
  gcn_pad_x_kernel<<<mapBlocks, blk, 0, stream>>>(x, xp, N);  // K=5 -> K=8, zero pad
  gcn_gemm_wmma_kernel<8, C32><<<gemmBlocks, blk, 0, stream>>>(xp, W1, h2, tilesM);
  if (remRows)
    gcn_gemm_tail_kernel<8, C32><<<4, blk, 0, stream>>>(xp, W1, h2, tilesM << 4, N);
  hipMemsetAsync(agg, 0, (size_t)N * C32 * sizeof(float), stream);
  gcn_aggregate32_kernel<<<waveEdgeBlocks, blk, 0, stream>>>(h2, dinv, src, dst, agg, E);
  gcn_selfloop32_kernel<<<nodeWaveBlocks, blk, 0, stream>>>(h2, dinv, agg, N);
  hipMemsetAsync(stats, 0, 64 * sizeof(float), stream);
  gcn_bn_stats_kernel<<<mapBlocks, blk, 0, stream>>>(agg, stats, N * C32);
  gcn_bn_finalize_kernel<<<1, 32, 0, stream>>>(stats, gamma1, beta1, N);
  gcn_bn_relu_kernel<<<mapBlocks, blk, 0, stream>>>(agg, stats, h, N * C32);

  // ---- layer 2: conv(h, W2) -> BN -> ReLU ----
  gcn_gemm_wmma_kernel<C32, C32><<<gemmBlocks, blk, 0, stream>>>(h, W2, h2, tilesM);
  if (remRows)
    gcn_gemm_tail_kernel<C32, C32><<<4, blk, 0, stream>>>(h, W2, h2, tilesM << 4, N);
  hipMemsetAsync(agg, 0, (size_t)N * C32 * sizeof(float), stream);
  gcn_aggregate32_kernel<<<waveEdgeBlocks, blk, 0, stream>>>(h2, dinv, src, dst, agg, E);
  gcn_selfloop32_kernel<<<nodeWaveBlocks, blk, 0, stream>>>(h2, dinv, agg, N);
  hipMemsetAsync(stats, 0, 64 * sizeof(float), stream);
  gcn_bn_stats_kernel<<<mapBlocks, blk, 0, stream>>>(agg, stats, N * C32);
  gcn_bn_finalize_kernel<<<1, 32, 0, stream>>>(stats, gamma2, beta2, N);
  gcn_bn_relu_kernel<<<mapBlocks, blk, 0, stream>>>(agg, stats, h, N * C32);

  // ---- layer 3: conv(h, W3) -> out ----
  gcn_gemv32_kernel<<<nodeWaveBlocks, blk, 0, stream>>>(h, W3, pre, N);
  hipMemsetAsync(out, 0, (size_t)N * sizeof(float), stream);
  gcn_aggregate1_kernel<<<edgeBlocks, blk, 0, stream>>>(pre, dinv, src, dst, out, E);
  gcn_selfloop1_kernel<<<mapBlocks, blk, 0, stream>>>(pre, dinv, b3, out, N);
}